// ManoModel_21981642620944
// MI455X (gfx1250) — compile-verified
//
#include <hip/hip_runtime.h>

typedef float v2f __attribute__((ext_vector_type(2)));
typedef float v8f __attribute__((ext_vector_type(8)));

#define Bsz 8192
#define Vn  778
#define NC  2334      // V*3
#define NBn 10

#define KTOT    152   // padded K: [1 | betas(10) | pad | pose_feature(135) | pad]
#define KSTEPS  38    // KTOT/4
#define KF_STRIDE 160
#define CSTRIDE 2352  // 49*48, padded column stride

// workspace offsets (floats)
// SD is stored K-PAIR interleaved: SDP[kp][c][i] = M[2*kp+i][c], kp<76
#define WS_SD 0
#define SD_SIZE (KTOT * CSTRIDE)              // 357504 (= 76*2352*2)
#define WS_JS (WS_SD + SD_SIZE)               // J_regressor @ shapedirs : 16*3*10
#define WS_JT (WS_JS + 480)                   // J_regressor @ v_template: 16*3
#define WS_KF (WS_JT + 48)                    // per-batch K feature rows: B*160
#define WS_A  (WS_KF + Bsz * KF_STRIDE)       // rel transforms: B*16*12

#define OUT_JT ((size_t)Bsz * Vn * 3)         // tail of d_out: J_transformed

// ------- setup: pack [v_template | shapedirs | posedirs], K-pair interleaved -------
__global__ void k_pack_sd(const float* vtempl, const float* shapedirs,
                          const float* posedirs, float* ws) {
  int idx = blockIdx.x * 256 + threadIdx.x;
  if (idx >= KTOT * CSTRIDE) return;
  // layout: ((kp*CSTRIDE + c)*2 + i), k = 2*kp + i
  int kp  = idx / (CSTRIDE * 2);
  int rem = idx % (CSTRIDE * 2);
  int c = rem >> 1, i = rem & 1;
  int k = 2 * kp + i;
  float val = 0.f;
  if (c < NC) {
    if (k == 0)                  val = vtempl[c];                     // constant-1 feature
    else if (k <= 10)            val = shapedirs[c * NBn + (k - 1)];  // (V,3,NB)
    else if (k >= 12 && k < 147) val = posedirs[(k - 12) * NC + c];   // (135, V*3)
  }
  ws[WS_SD + idx] = val;
}

// ---------------- setup: fold J_regressor into shapedirs / template ----------------
__global__ void k_joint_reg(const float* Jreg, const float* shapedirs,
                            const float* vtempl, float* ws) {
  int idx = blockIdx.x * 256 + threadIdx.x;
  if (idx >= 528) return;
  if (idx < 480) {                       // JS[j][c][l]
    int l = idx % 10, jc = idx / 10, c = jc % 3, j = jc / 3;
    float s = 0.f;
    for (int v = 0; v < Vn; ++v) s += Jreg[j * Vn + v] * shapedirs[(v * 3 + c) * NBn + l];
    ws[WS_JS + idx] = s;
  } else {                               // Jt[j][c]
    int t = idx - 480, c = t % 3, j = t / 3;
    float s = 0.f;
    for (int v = 0; v < Vn; ++v) s += Jreg[j * Vn + v] * vtempl[v * 3 + c];
    ws[WS_JT + t] = s;
  }
}

// ---------------- per-batch: Rodrigues, K-features, rigid chain --------------------
__global__ __launch_bounds__(128) void k_rig(const float* betas, const float* pose,
                                             float* ws, float* out) {
  int lane = threadIdx.x & 31;
  int b = blockIdx.x * 4 + (threadIdx.x >> 5);
  int j = lane & 15;

  float* Kf = ws + WS_KF + (size_t)b * KF_STRIDE;

  // housekeeping slots: [0]=1, [1..10]=betas, [11]=0, [147..159]=0
  for (int idx = lane; idx < KF_STRIDE; idx += 32) {
    if (idx >= 12 && idx < 147) continue;
    float val = 0.f;
    if (idx == 0) val = 1.f;
    else if (idx <= 10) val = betas[b * NBn + idx - 1];
    Kf[idx] = val;
  }

  // Rodrigues for joint j
  float px = pose[b * 48 + j * 3 + 0];
  float py = pose[b * 48 + j * 3 + 1];
  float pz = pose[b * 48 + j * 3 + 2];
  float ax = px + 1e-8f, ay = py + 1e-8f, az = pz + 1e-8f;
  float angle = sqrtf(ax * ax + ay * ay + az * az);
  float inv = 1.f / angle;
  float rx = px * inv, ry = py * inv, rz = pz * inv;
  float sn = sinf(angle), cm = 1.f - cosf(angle);
  float R00 = 1.f + cm * (-(ry * ry + rz * rz));
  float R01 = -sn * rz + cm * (rx * ry);
  float R02 =  sn * ry + cm * (rx * rz);
  float R10 =  sn * rz + cm * (rx * ry);
  float R11 = 1.f + cm * (-(rx * rx + rz * rz));
  float R12 = -sn * rx + cm * (ry * rz);
  float R20 = -sn * ry + cm * (rx * rz);
  float R21 =  sn * rx + cm * (ry * rz);
  float R22 = 1.f + cm * (-(rx * rx + ry * ry));

  // pose_feature = R - I for joints 1..15
  if (lane >= 1 && lane < 16) {
    float* pf = Kf + 12 + (j - 1) * 9;
    pf[0] = R00 - 1.f; pf[1] = R01;       pf[2] = R02;
    pf[3] = R10;       pf[4] = R11 - 1.f; pf[5] = R12;
    pf[6] = R20;       pf[7] = R21;       pf[8] = R22 - 1.f;
  }

  // rest joint location for j: Jt + JS @ betas
  float jl[3];
#pragma unroll
  for (int c = 0; c < 3; ++c) {
    float s = ws[WS_JT + j * 3 + c];
    const float* js = ws + WS_JS + (j * 3 + c) * NBn;
#pragma unroll
    for (int l = 0; l < NBn; ++l) s += js[l] * betas[b * NBn + l];
    jl[c] = s;
  }

  // kinematic tree: parents = [0,0,1,2,0,4,5,0,7,8,0,10,11,0,13,14]
  int par   = (j == 0) ? 0 : ((j % 3 == 1) ? 0 : j - 1);
  int depth = (j == 0) ? 0 : ((j - 1) % 3) + 1;

  float pj0 = __shfl(jl[0], par, 32);
  float pj1 = __shfl(jl[1], par, 32);
  float pj2 = __shfl(jl[2], par, 32);
  float t0 = (j == 0) ? jl[0] : jl[0] - pj0;
  float t1 = (j == 0) ? jl[1] : jl[1] - pj1;
  float t2 = (j == 0) ? jl[2] : jl[2] - pj2;

  float G[12] = {R00, R01, R02, t0, R10, R11, R12, t1, R20, R21, R22, t2};

  // depth-3 tree -> 3 shuffle-compose steps
#pragma unroll
  for (int d = 1; d <= 3; ++d) {
    float P[12];
#pragma unroll
    for (int q = 0; q < 12; ++q) P[q] = __shfl(G[q], par, 32);
    if (depth == d) {
      float n[12];
#pragma unroll
      for (int r = 0; r < 3; ++r) {
#pragma unroll
        for (int cc = 0; cc < 4; ++cc) {
          float acc = (cc == 3) ? P[r * 4 + 3] : 0.f;
#pragma unroll
          for (int m = 0; m < 3; ++m) acc += P[r * 4 + m] * G[m * 4 + cc];
          n[r * 4 + cc] = acc;
        }
      }
#pragma unroll
      for (int q = 0; q < 12; ++q) G[q] = n[q];
    }
  }

  if (lane < 16) {
    // posed joints
    float* jt = out + OUT_JT + ((size_t)b * 16 + j) * 3;
    jt[0] = G[3]; jt[1] = G[7]; jt[2] = G[11];
    // relative transform: t' = t - R * jl
    float* A = ws + WS_A + ((size_t)b * 16 + j) * 12;
#pragma unroll
    for (int r = 0; r < 3; ++r) {
      A[r * 4 + 0] = G[r * 4 + 0];
      A[r * 4 + 1] = G[r * 4 + 1];
      A[r * 4 + 2] = G[r * 4 + 2];
      A[r * 4 + 3] = G[r * 4 + 3] -
                     (G[r * 4 + 0] * jl[0] + G[r * 4 + 1] * jl[1] + G[r * 4 + 2] * jl[2]);
    }
  }
}

// ---------------- main: WMMA v_posed GEMM + LBS blend ------------------------------
__global__ __launch_bounds__(128) void k_verts(const float* lbsw, const float* ws,
                                               float* out) {
  __shared__ float sW[16][16];        // lbs weights for 16 verts
  __shared__ float sA[4][16][192];    // per-wave 16 batches x (16 joints x 12)
  __shared__ float sVP[4][16][48];    // per-wave v_posed tile: 16 b x 48 cols

  int tid = threadIdx.x;
  int lane = tid & 31, w = tid >> 5;
  int vt = blockIdx.x, by = blockIdx.y;
  int batchBase = by * 64 + w * 16;

  // stage lbs weights
  for (int i = tid; i < 256; i += 128) {
    int vl = i >> 4, j = i & 15, v = vt * 16 + vl;
    sW[vl][j] = (v < Vn) ? lbsw[v * 16 + j] : 0.f;
  }
  // stage rel transforms for this block's 64 batches
  const float* Aws = ws + WS_A;
  float* sAf = &sA[0][0][0];
  for (int i = tid; i < 64 * 192; i += 128)
    sAf[i] = Aws[((size_t)by * 64 + i / 192) * 192 + (i % 192)];

  // WMMA f32 16x16x4: M=16 batches, N=16 cols, K=152 (38 steps), 3 column groups.
  // SD is k-pair interleaved -> each B fragment is ONE aligned b64 load.
  const float* Kf = ws + WS_KF;
  const float* SD = ws + WS_SD;
  int half = lane >> 4, l16 = lane & 15;
  const float* kfRow = Kf + (size_t)(batchBase + l16) * KF_STRIDE + half * 2;
  int colBase = vt * 48 + l16;

  v8f acc0 = {0.f, 0.f, 0.f, 0.f, 0.f, 0.f, 0.f, 0.f};
  v8f acc1 = {0.f, 0.f, 0.f, 0.f, 0.f, 0.f, 0.f, 0.f};
  v8f acc2 = {0.f, 0.f, 0.f, 0.f, 0.f, 0.f, 0.f, 0.f};

  for (int ks = 0; ks < KSTEPS; ++ks) {
    v2f a = *(const v2f*)(kfRow + ks * 4);
    const v2f* sd = (const v2f*)(SD + ((size_t)(ks * 2 + half) * CSTRIDE + colBase) * 2);
    v2f b0 = sd[0];
    acc0 = __builtin_amdgcn_wmma_f32_16x16x4_f32(false, a, false, b0, (short)0, acc0, false, false);
    v2f b1 = sd[16];
    acc1 = __builtin_amdgcn_wmma_f32_16x16x4_f32(false, a, false, b1, (short)0, acc1, false, false);
    v2f b2 = sd[32];
    acc2 = __builtin_amdgcn_wmma_f32_16x16x4_f32(false, a, false, b2, (short)0, acc2, false, false);
  }

  // spill v_posed tile to LDS (C/D layout: VGPR r -> M=r (lanes 0-15), M=r+8 (lanes 16-31))
  int mBase = half * 8;
#pragma unroll
  for (int r = 0; r < 8; ++r) {
    sVP[w][mBase + r][l16]      = acc0[r];
    sVP[w][mBase + r][16 + l16] = acc1[r];
    sVP[w][mBase + r][32 + l16] = acc2[r];
  }
  __syncthreads();

  // LBS blend: verts[b,v,p] = sum_j w[v,j] * (A[b,j,p,:3] . vp + A[b,j,p,3])
  for (int i = 0; i < 8; ++i) {
    int p = lane + 32 * i;
    int bl = p >> 4, vl = p & 15;
    float x = sVP[w][bl][vl * 3 + 0];
    float y = sVP[w][bl][vl * 3 + 1];
    float z = sVP[w][bl][vl * 3 + 2];
    float o0 = 0.f, o1 = 0.f, o2 = 0.f;
    const float* Ab = &sA[w][bl][0];
#pragma unroll
    for (int j = 0; j < 16; ++j) {
      float wj = sW[vl][j];
      const float* a = Ab + j * 12;
      o0 += wj * (a[0] * x + a[1] * y + a[2]  * z + a[3]);
      o1 += wj * (a[4] * x + a[5] * y + a[6]  * z + a[7]);
      o2 += wj * (a[8] * x + a[9] * y + a[10] * z + a[11]);
    }
    int v = vt * 16 + vl;
    if (v < Vn) {
      size_t o = ((size_t)(batchBase + bl) * Vn + v) * 3;
      out[o + 0] = o0; out[o + 1] = o1; out[o + 2] = o2;
    }
  }
}

extern "C" void kernel_launch(void* const* d_in, const int* in_sizes, int n_in,
                              void* d_out, int out_size, void* d_ws, size_t ws_size,
                              hipStream_t stream) {
  const float* betas     = (const float*)d_in[0];
  const float* pose      = (const float*)d_in[1];
  const float* vtempl    = (const float*)d_in[2];
  const float* shapedirs = (const float*)d_in[3];
  const float* posedirs  = (const float*)d_in[4];
  const float* Jreg      = (const float*)d_in[5];
  const float* lbsw      = (const float*)d_in[6];
  // parents (d_in[7]) is a compile-time-known constant tree; hardcoded in k_rig.

  float* ws  = (float*)d_ws;
  float* out = (float*)d_out;

  k_pack_sd <<<(KTOT * CSTRIDE + 255) / 256, 256, 0, stream>>>(vtempl, shapedirs, posedirs, ws);
  k_joint_reg<<<(528 + 255) / 256,            256, 0, stream>>>(Jreg, shapedirs, vtempl, ws);
  k_rig     <<<Bsz / 4,                        128, 0, stream>>>(betas, pose, ws, out);
  k_verts   <<<dim3((Vn + 15) / 16, Bsz / 64), 128, 0, stream>>>(lbsw, ws, out);
}